// BoundleAdjustment_17815524344322
// MI455X (gfx1250) — compile-verified
//
#include <hip/hip_runtime.h>

typedef float v2f __attribute__((ext_vector_type(2)));
typedef float v8f __attribute__((ext_vector_type(8)));

// ds_swizzle_b32 group-of-32 xor patterns: offset = (xor_mask<<10) | 0x1F
#define SWZ_XOR16 0x401F
#define SWZ_XOR8  0x201F
#define SWZ_XOR4  0x101F
#define SWZ_XOR2  0x081F
#define SWZ_XOR1  0x041F
#define SWZ_ADD(v, pat) \
  ((v) + __int_as_float(__builtin_amdgcn_ds_swizzle(__float_as_int(v), (pat))))

// Full-wave32 f32 sum using 5 single-instruction ds_swizzle lane exchanges.
// All 32 lanes must be active (EXEC all-ones) at the call site.
__device__ __forceinline__ float wave_reduce_sum(float v) {
  v = SWZ_ADD(v, SWZ_XOR16);
  v = SWZ_ADD(v, SWZ_XOR8);
  v = SWZ_ADD(v, SWZ_XOR4);
  v = SWZ_ADD(v, SWZ_XOR2);
  v = SWZ_ADD(v, SWZ_XOR1);
  return v;
}

// One wave32 per frame (grid-strided). Lane = camera index (C=24 active lanes).
// Streams pole2d/mask exactly once; camera params live in registers per lane.
__global__ void __launch_bounds__(256)
frame_loss_kernel(const float* __restrict__ pole3d,
                  const float* __restrict__ pole2d,
                  const float* __restrict__ Kmat,
                  const float* __restrict__ dist,
                  const float* __restrict__ Rmat,
                  const float* __restrict__ tvec,
                  const int*   __restrict__ mask,
                  float* __restrict__ frame_loss,
                  int M, int C, int Mpad) {
  const int lane   = threadIdx.x & 31;
  const int wave   = blockIdx.x * (blockDim.x >> 5) + (threadIdx.x >> 5);
  const int nwaves = gridDim.x * (blockDim.x >> 5);
  const bool act   = lane < C;

  // Per-lane camera constants (tiny, L2-resident; loaded once per wave).
  float fx = 0.f, fy = 0.f, u0 = 0.f, v0 = 0.f;
  float k1 = 0.f, k2 = 0.f, p1 = 0.f, p2 = 0.f, k3 = 0.f;
  float r00 = 0.f, r01 = 0.f, r02 = 0.f;
  float r10 = 0.f, r11 = 0.f, r12 = 0.f;
  float r20 = 0.f, r21 = 0.f, r22 = 0.f;
  float t0 = 0.f, t1 = 0.f, t2 = 0.f;
  if (act) {
    const float* Kc = Kmat + lane * 9;
    fx = Kc[0]; fy = Kc[4]; u0 = Kc[2]; v0 = Kc[5];
    const float* dc = dist + lane * 5;
    k1 = dc[0]; k2 = dc[1]; p1 = dc[2]; p2 = dc[3]; k3 = dc[4];
    const float* Rc = Rmat + lane * 9;
    r00 = Rc[0]; r01 = Rc[1]; r02 = Rc[2];
    r10 = Rc[3]; r11 = Rc[4]; r12 = Rc[5];
    r20 = Rc[6]; r21 = Rc[7]; r22 = Rc[8];
    const float* tc = tvec + lane * 3;
    t0 = tc[0]; t1 = tc[1]; t2 = tc[2];
  }

  const int stride2d = C * 6;  // floats per frame in pole2d
  for (int m = wave; m < Mpad; m += nwaves) {
    float loss = 0.0f;
    if (m < M) {
      const int mn = m + nwaves;  // prefetch next frame's detections
      if (act && mn < M)
        __builtin_prefetch(pole2d + (size_t)mn * stride2d + lane * 6, 0, 1);

      const float* X = pole3d + (size_t)m * 9;  // broadcast load
      const float x00 = X[0], x01 = X[1], x02 = X[2];
      const float x10 = X[3], x11 = X[4], x12 = X[5];
      const float x20 = X[6], x21 = X[7], x22 = X[8];

      const int mk = act ? mask[(size_t)m * C + lane] : 0;
      float num = 0.0f;
      if (act) {
        const float* det = pole2d + (size_t)m * stride2d + lane * 6;
        float errsum = 0.0f;
#pragma unroll
        for (int p = 0; p < 3; ++p) {
          const float X0 = (p == 0) ? x00 : ((p == 1) ? x10 : x20);
          const float X1 = (p == 0) ? x01 : ((p == 1) ? x11 : x21);
          const float X2 = (p == 0) ? x02 : ((p == 1) ? x12 : x22);
          // camera frame
          const float c0 = fmaf(r00, X0, fmaf(r01, X1, fmaf(r02, X2, t0)));
          const float c1 = fmaf(r10, X0, fmaf(r11, X1, fmaf(r12, X2, t1)));
          const float c2 = fmaf(r20, X0, fmaf(r21, X1, fmaf(r22, X2, t2)));
          const float invz = 1.0f / c2;
          const float a0 = c0 * invz, a1 = c1 * invz;
          const float r2 = a0 * a0 + a1 * a1;
          const float radial = 1.0f + r2 * (k1 + r2 * (k2 + r2 * k3));
          const float xu = a0 * radial + 2.0f * p1 * a0 * a1 + p2 * (r2 + 2.0f * a0 * a0);
          const float yu = a1 * radial + p1 * (r2 + 2.0f * a1 * a1) + 2.0f * p2 * a0 * a1;
          const float du = det[p * 2 + 0] - fmaf(fx, xu, u0);
          const float dv = det[p * 2 + 1] - fmaf(fy, yu, v0);
          errsum += sqrtf(du * du + dv * dv);
        }
        num = (float)mk * errsum * (1.0f / 3.0f);
      }
      // visible-camera count via wave ballot (mask is 0/1): no float tree needed
      const unsigned vis = __builtin_amdgcn_ballot_w32(mk != 0);
      num = wave_reduce_sum(num);                // EXEC all-ones here (reconverged)
      loss = num / (float)__popc(vis);           // >=1 visible camera guaranteed
    }
    if (lane == 0) frame_loss[m] = loss;  // zero for padded tail
  }
}

// Matrix-pipe reduction: D = A(16x4) * ones(4x16) + C accumulates 64-element
// sums per v_wmma_f32_16x16x4_f32. Sum of all 256 D elements = 16 * S.
// Software-pipelined: next tile's global_load_b64 is in flight behind the WMMA.
__global__ void __launch_bounds__(32)
reduce64_wmma_kernel(const float* __restrict__ vals,
                     float* __restrict__ partials, int numTiles) {
  const int lane = threadIdx.x;  // one wave32 per block, EXEC all-ones
  v8f acc = {0.f, 0.f, 0.f, 0.f, 0.f, 0.f, 0.f, 0.f};
  const v2f ones = {1.0f, 1.0f};
  const int t0i = blockIdx.x;
  if (t0i < numTiles) {  // scalar-uniform branch: EXEC stays all-ones
    v2f cur = *(const v2f*)(vals + (size_t)t0i * 64 + lane * 2);
    for (int n = t0i + gridDim.x; n < numTiles; n += gridDim.x) {
      const v2f nxt = *(const v2f*)(vals + (size_t)n * 64 + lane * 2);
      acc = __builtin_amdgcn_wmma_f32_16x16x4_f32(
          false, cur, false, ones, (short)0, acc, false, false);
      cur = nxt;
    }
    acc = __builtin_amdgcn_wmma_f32_16x16x4_f32(
        false, cur, false, ones, (short)0, acc, false, false);
  }
  float s = acc[0] + acc[1] + acc[2] + acc[3] + acc[4] + acc[5] + acc[6] + acc[7];
  s = wave_reduce_sum(s);
  if (lane == 0) partials[blockIdx.x] = s * (1.0f / 16.0f);
}

__global__ void __launch_bounds__(32)
finalize_kernel(const float* __restrict__ partials, float* __restrict__ out,
                int n, float invM) {
  const int lane = threadIdx.x;
  float s = 0.f;
  for (int i = lane; i < n; i += 32) s += partials[i];
  s = wave_reduce_sum(s);
  if (lane == 0) out[0] = s * invM;
}

extern "C" void kernel_launch(void* const* d_in, const int* in_sizes, int n_in,
                              void* d_out, int out_size, void* d_ws, size_t ws_size,
                              hipStream_t stream) {
  (void)n_in; (void)out_size; (void)ws_size;
  const float* pole3d = (const float*)d_in[0];
  const float* pole2d = (const float*)d_in[1];
  const float* K      = (const float*)d_in[2];
  const float* dist   = (const float*)d_in[3];
  const float* R      = (const float*)d_in[4];
  const float* t      = (const float*)d_in[5];
  // d_in[6] = pole[d1,d2]: unused — wand terms have zero weight in forward()
  const int*   mask   = (const int*)d_in[7];

  const int M    = in_sizes[0] / 9;   // pole3d is [M,3,3]
  const int C    = in_sizes[3] / 5;   // dist is [C,5]
  const int Mpad = ((M + 63) / 64) * 64;

  float* frame_loss = (float*)d_ws;        // [Mpad]
  float* partials   = frame_loss + Mpad;   // [G]

  const int G = 128;  // stage-A single-wave blocks
  frame_loss_kernel<<<1024, 256, 0, stream>>>(pole3d, pole2d, K, dist, R, t,
                                              mask, frame_loss, M, C, Mpad);
  reduce64_wmma_kernel<<<G, 32, 0, stream>>>(frame_loss, partials, Mpad / 64);
  finalize_kernel<<<1, 32, 0, stream>>>(partials, (float*)d_out, G,
                                        1.0f / (float)M);
}